// DACRVQBottleneck_44298292691485
// MI455X (gfx1250) — compile-verified
//
#include <hip/hip_runtime.h>
#include <hip/hip_bf16.h>

// ---------------------------------------------------------------------------
// RVQ (DAC ResidualVectorQuantize) fused forward for MI455X (gfx1250).
//
// Roofline: ~21.7 GFLOP total, ~110 MB min HBM traffic -> ~4.7us mem floor at
// 23.3 TB/s; ~200 FLOP/B intensity forces the K=1024 codeword search through
// the matrix pipes. We use V_WMMA_F32_16X16X4_F32 (fp32 end-to-end, matching
// the fp32 reference) for in_proj, codeword dot-products, and out_proj.
// All 9 codebooks are fused per 256-timestep tile: x read once, z_q written
// once, residual chain in LDS (wave-private columns, store-only updates),
// codebook fetch for iteration i+1 pipelined behind iteration i's WMMA loop.
// ---------------------------------------------------------------------------

typedef __attribute__((ext_vector_type(2))) float v2f;
typedef __attribute__((ext_vector_type(8))) float v8f;

#define NCB 9
#define B_  4
#define D_  64
#define T_  32768
#define K_  1024
#define CD_ 8

#define TILE_T 256           // timesteps per block (16 waves x 16)
#define NTHR   512
#define RS     260           // residual LDS row stride (2*RS mod 64 = 8, no conflict)
#define ZS     260           // z_e LDS row stride
#define CBS    9             // normalized-codebook LDS row stride (coprime w/ 64)

// d_out layout (flat, reference return order, all as float):
//   z_q   [4,64,32768]  -> offset 0,        8388608 elems
//   codes [4, 9,32768]  -> offset 8388608,  1179648 elems
//   lat   [4,72,32768]  -> offset 9568256,  9437184 elems
//   commit, cb_loss     -> offsets 19005440, 19005441
#define CODES_OFF 8388608
#define LAT_OFF   9568256
#define LOSS_OFF  19005440

__global__ void rvq_zero_losses(float* __restrict__ loss) {
  loss[0] = 0.f;
  loss[1] = 0.f;
}

__launch_bounds__(NTHR)
__global__ void rvq_fused_kernel(const float* __restrict__ x,
                                 const float* __restrict__ W_in,
                                 const float* __restrict__ b_in,
                                 const float* __restrict__ codebooks,
                                 const float* __restrict__ W_out,
                                 const float* __restrict__ b_out,
                                 float* __restrict__ zq_out,
                                 float* __restrict__ codes_out,
                                 float* __restrict__ lat_out,
                                 float* __restrict__ loss_out)
{
  __shared__ float res [64 * RS];      // residual tile [d][t] (wave-private cols)
  __shared__ float cbn [K_ * CBS];     // L2-normalized codebook rows
  __shared__ float ze  [CD_ * ZS];     // raw z_e tile [c][t]
  __shared__ float rnv [TILE_T];       // 1/max(||z_e||,eps) per column
  __shared__ int   idxs[TILE_T];       // argmin index per column
  __shared__ float winp[16 * 64];      // W_in padded to 16 rows
  __shared__ float wout[64 * CD_];
  __shared__ float binp[16];
  __shared__ float boutp[64];

  const int tid  = threadIdx.x;
  const int lane = tid & 31;
  const int wave = tid >> 5;
  const int half = lane >> 4;          // wave32 lane half (WMMA layout split)
  const int lm   = lane & 15;
  const int blk  = blockIdx.x;
  const int b    = blk >> 7;           // 128 t-tiles per batch
  const int tb   = (blk & 127) * TILE_T;
  const int t0w  = wave * 16;          // wave's column base within tile

  // ---- x tile in C-layout registers; seed residual LDS (store-only) -----
  // slot (mt, v) on lane (half,lm): d = mt*16 + v + 8*half, t = t0w + lm.
  float xreg[4][8];
  v8f   zqacc[4];                      // cumulative out-proj == z_q
  #pragma unroll
  for (int mt = 0; mt < 4; ++mt) {
    zqacc[mt] = (v8f){0.f,0.f,0.f,0.f,0.f,0.f,0.f,0.f};
    #pragma unroll
    for (int v = 0; v < 8; ++v) {
      int d = mt * 16 + v + 8 * half;
      float xv = x[(size_t)b * D_ * T_ + (size_t)d * T_ + tb + t0w + lm];
      xreg[mt][v] = xv;
      res[d * RS + t0w + lm] = xv;     // wave-private columns: no block sync
    }
  }
  float loss_acc = 0.f;

  // ---- software-pipelined codebook staging (2 rows / thread) ------------
  const int row0 = tid * 2;
  float cbreg[2][CD_];
  #pragma unroll
  for (int rr = 0; rr < 2; ++rr)
    #pragma unroll
    for (int c = 0; c < CD_; ++c)
      cbreg[rr][c] = codebooks[(size_t)(row0 + rr) * CD_ + c];  // codebook 0

  for (int i = 0; i < NCB; ++i) {
    // normalize the pre-fetched codebook rows into LDS
    #pragma unroll
    for (int rr = 0; rr < 2; ++rr) {
      float s = 0.f;
      #pragma unroll
      for (int c = 0; c < CD_; ++c) s += cbreg[rr][c] * cbreg[rr][c];
      float r = 1.f / fmaxf(sqrtf(s), 1e-12f);
      #pragma unroll
      for (int c = 0; c < CD_; ++c)
        cbn[(row0 + rr) * CBS + c] = cbreg[rr][c] * r;
    }
    // stage this codebook's weights (small, L2-hot)
    for (int r = tid; r < 16 * 64; r += NTHR) {
      int c = r >> 6, d = r & 63;
      winp[r] = (c < CD_) ? W_in[((size_t)i * CD_ + c) * 64 + d] : 0.f;
    }
    for (int r = tid; r < 64 * CD_; r += NTHR)
      wout[r] = W_out[(size_t)i * 64 * CD_ + r];
    if (tid < 16) binp[tid]  = (tid < CD_) ? b_in[i * CD_ + tid] : 0.f;
    if (tid < 64) boutp[tid] = b_out[i * 64 + tid];
    __syncthreads();

    // issue next codebook's loads now; consumed after the WMMA loops, so the
    // L2 latency hides behind ~130 WMMAs + argmax VALU work.
    if (i < NCB - 1) {
      #pragma unroll
      for (int rr = 0; rr < 2; ++rr)
        #pragma unroll
        for (int c = 0; c < CD_; ++c)
          cbreg[rr][c] = codebooks[((size_t)(i + 1) * K_ + row0 + rr) * CD_ + c];
    }

    // ---- in_proj: z_e[c,t] = W_in[c,:] . res[:,t]  (WMMA, K=64 in 16x4) --
    v8f zea = (v8f){0.f,0.f,0.f,0.f,0.f,0.f,0.f,0.f};
    for (int k = 0; k < 16; ++k) {
      int ka = 4 * k + 2 * half;
      v2f a, bf;
      a.x  = winp[lm * 64 + ka];
      a.y  = winp[lm * 64 + ka + 1];
      bf.x = res[(ka)     * RS + t0w + lm];
      bf.y = res[(ka + 1) * RS + t0w + lm];
      zea = __builtin_amdgcn_wmma_f32_16x16x4_f32(false, a, false, bf,
                                                  (short)0, zea, false, false);
    }
    #pragma unroll
    for (int v = 0; v < 8; ++v) zea[v] += binp[v + 8 * half];

    // slot (v, half==0): c=v, t=t0w+lm. Stage z_e + latents + column rnorm.
    if (half == 0) {
      float s = 0.f;
      #pragma unroll
      for (int v = 0; v < 8; ++v) {
        float z = zea[v];
        ze[v * ZS + t0w + lm] = z;
        lat_out[(size_t)b * (NCB * CD_) * T_ + (size_t)(i * CD_ + v) * T_ + tb + t0w + lm] = z;
        s += z * z;
      }
      rnv[t0w + lm] = 1.f / fmaxf(sqrtf(s), 1e-12f);
    }

    // ---- nearest codeword: argmax over K of enc.cbk (both unit-norm) ----
    // dist = |enc|^2 - 2 enc.cbk + |cbk|^2 = 2 - 2 enc.cbk after normalize.
    float r_t = rnv[t0w + lm];
    v2f ad0, ad1;                        // A fragments: M=t, K=c (2 chunks)
    ad0.x = ze[(0 + 2 * half) * ZS + t0w + lm] * r_t;
    ad0.y = ze[(1 + 2 * half) * ZS + t0w + lm] * r_t;
    ad1.x = ze[(4 + 2 * half) * ZS + t0w + lm] * r_t;
    ad1.y = ze[(5 + 2 * half) * ZS + t0w + lm] * r_t;

    float bv[8]; int bi[8];
    #pragma unroll
    for (int v = 0; v < 8; ++v) { bv[v] = -3.0e38f; bi[v] = 0; }

    for (int kt = 0; kt < 64; ++kt) {
      int cw = kt * 16 + lm;             // this lane's codeword column
      int cb0 = cw * CBS + 2 * half;
      v2f b0, b1;
      b0.x = cbn[cb0];     b0.y = cbn[cb0 + 1];
      b1.x = cbn[cb0 + 4]; b1.y = cbn[cb0 + 5];
      v8f dacc = (v8f){0.f,0.f,0.f,0.f,0.f,0.f,0.f,0.f};
      dacc = __builtin_amdgcn_wmma_f32_16x16x4_f32(false, ad0, false, b0,
                                                   (short)0, dacc, false, false);
      dacc = __builtin_amdgcn_wmma_f32_16x16x4_f32(false, ad1, false, b1,
                                                   (short)0, dacc, false, false);
      #pragma unroll
      for (int v = 0; v < 8; ++v) {      // slot: t = v + 8*half, k = cw
        bool u = dacc[v] > bv[v];
        bv[v] = u ? dacc[v] : bv[v];
        bi[v] = u ? cw      : bi[v];
      }
    }

    // reduce argmax across the 16 lanes of each half (masks<16 stay in-half)
    #pragma unroll
    for (int v = 0; v < 8; ++v) {
      #pragma unroll
      for (int off = 8; off >= 1; off >>= 1) {
        float ov = __shfl_xor(bv[v], off, 32);
        int   oi = __shfl_xor(bi[v], off, 32);
        bool take = (ov > bv[v]) || (ov == bv[v] && oi < bi[v]);
        bv[v] = take ? ov : bv[v];
        bi[v] = take ? oi : bi[v];
      }
      if (lm == 0) {
        int tl = t0w + v + 8 * half;
        idxs[tl] = bi[v];
        codes_out[(size_t)b * NCB * T_ + (size_t)i * T_ + tb + tl] = (float)bi[v];
      }
    }

    // ---- gather z_q_i (raw codebook rows, L2-resident) ------------------
    int myidx = idxs[t0w + lm];
    const float* cbrow = codebooks + ((size_t)i * K_ + myidx) * CD_;
    v2f bq0, bq1;                        // B fragments: K=c, N=t
    bq0.x = cbrow[0 + 2 * half]; bq0.y = cbrow[1 + 2 * half];
    bq1.x = cbrow[4 + 2 * half]; bq1.y = cbrow[5 + 2 * half];

    // commit/codebook loss (identical in forward): sum (z_e - z_q_i)^2
    if (half == 0) {
      #pragma unroll
      for (int v = 0; v < 8; ++v) {
        float d = zea[v] - cbrow[v];
        loss_acc += d * d;
      }
    }

    // ---- out_proj + residual/z_q update (straight-through: z_st = z_q_i) -
    // residual = x - cumulative(out): store-only LDS update, no RMW chain.
    #pragma unroll
    for (int mt = 0; mt < 4; ++mt) {
      int dbase = mt * 16 + lm;
      v2f aw0, aw1;
      aw0.x = wout[dbase * CD_ + 0 + 2 * half];
      aw0.y = wout[dbase * CD_ + 1 + 2 * half];
      aw1.x = wout[dbase * CD_ + 4 + 2 * half];
      aw1.y = wout[dbase * CD_ + 5 + 2 * half];
      v8f oacc = (v8f){0.f,0.f,0.f,0.f,0.f,0.f,0.f,0.f};
      oacc = __builtin_amdgcn_wmma_f32_16x16x4_f32(false, aw0, false, bq0,
                                                   (short)0, oacc, false, false);
      oacc = __builtin_amdgcn_wmma_f32_16x16x4_f32(false, aw1, false, bq1,
                                                   (short)0, oacc, false, false);
      #pragma unroll
      for (int v = 0; v < 8; ++v) {      // slot: d = mt*16+v+8*half, t = lm
        zqacc[mt][v] += oacc[v] + boutp[mt * 16 + v + 8 * half];
        res[(mt * 16 + v + 8 * half) * RS + t0w + lm]
            = xreg[mt][v] - zqacc[mt][v];
      }
    }
    __syncthreads();   // all waves done with cbn/winp/wout before restage
  }

  // ---- epilogue: z_q store + loss reduction -----------------------------
  #pragma unroll
  for (int mt = 0; mt < 4; ++mt)
    #pragma unroll
    for (int v = 0; v < 8; ++v)
      zq_out[(size_t)b * D_ * T_ + (size_t)(mt * 16 + v + 8 * half) * T_ + tb + t0w + lm]
          = zqacc[mt][v];

  float ls = loss_acc;
  #pragma unroll
  for (int off = 16; off >= 1; off >>= 1) ls += __shfl_xor(ls, off, 32);
  if (lane == 0) {
    const float scale = 1.0f / ((float)B_ * CD_ * T_ * NCB);  // mean, /n
    atomicAdd(&loss_out[0], ls * scale);   // commit
    atomicAdd(&loss_out[1], ls * scale);   // codebook loss (same forward value)
  }
}

extern "C" void kernel_launch(void* const* d_in, const int* in_sizes, int n_in,
                              void* d_out, int out_size, void* d_ws, size_t ws_size,
                              hipStream_t stream) {
  (void)in_sizes; (void)n_in; (void)d_ws; (void)ws_size; (void)out_size;
  const float* x         = (const float*)d_in[0];
  const float* W_in      = (const float*)d_in[1];
  const float* b_in      = (const float*)d_in[2];
  const float* codebooks = (const float*)d_in[3];
  const float* W_out     = (const float*)d_in[4];
  const float* b_out     = (const float*)d_in[5];

  float* out   = (float*)d_out;
  float* zq    = out;
  float* codes = out + CODES_OFF;
  float* lat   = out + LAT_OFF;
  float* loss  = out + LOSS_OFF;

  rvq_zero_losses<<<dim3(1), dim3(1), 0, stream>>>(loss);
  rvq_fused_kernel<<<dim3(B_ * (T_ / TILE_T)), dim3(NTHR), 0, stream>>>(
      x, W_in, b_in, codebooks, W_out, b_out, zq, codes, lat, loss);
}